// InvoiceGCN_10402410791478
// MI455X (gfx1250) — compile-verified
//
#include <hip/hip_runtime.h>
#include <cstdint>
#include <cstddef>

typedef __attribute__((ext_vector_type(2))) float v2f;
typedef __attribute__((ext_vector_type(8))) float v8f;

// ---------------------------------------------------------------- utilities

__global__ void zero_f32_kernel(float* __restrict__ p, size_t n) {
  size_t i = (size_t)blockIdx.x * blockDim.x + threadIdx.x;
  size_t stride = (size_t)gridDim.x * blockDim.x;
  for (; i < n; i += stride) p[i] = 0.0f;
}

__global__ void negcopy_kernel(float* __restrict__ dst, const float* __restrict__ src, size_t n) {
  size_t i = (size_t)blockIdx.x * blockDim.x + threadIdx.x;
  size_t stride = (size_t)gridDim.x * blockDim.x;
  for (; i < n; i += stride) dst[i] = -src[i];
}

// ------------------------------------------------- degree / norm precompute

__global__ void deg_scatter_kernel(const long long* __restrict__ row,
                                   const float* __restrict__ w,
                                   float* __restrict__ deg, int E) {
  int e = blockIdx.x * blockDim.x + threadIdx.x;
  if (e < E) atomicAdd(&deg[row[e]], w[e]);
}

__global__ void norm_kernel(const long long* __restrict__ row,
                            const long long* __restrict__ col,
                            const float* __restrict__ w,
                            const float* __restrict__ deg,
                            float* __restrict__ nrm, int E) {
  int e = blockIdx.x * blockDim.x + threadIdx.x;
  if (e < E) {
    float dr = deg[row[e]];
    float dc = deg[col[e]];
    float ir = dr > 0.0f ? rsqrtf(dr) : 0.0f;
    float ic = dc > 0.0f ? rsqrtf(dc) : 0.0f;
    nrm[e] = -ir * w[e] * ic;
  }
}

// -------------------------------------------------------- SpMM (scatter-add)
// out[row[e], f..f+3] += scale * norm[e] * X[col[e], f..f+3]
// One thread handles 4 consecutive features: b128 gather + 4 f32 atomics,
// edge metadata (row/col/norm) loaded once per 4 features.

__global__ void spmm_scatter4_kernel(const long long* __restrict__ row,
                                     const long long* __restrict__ col,
                                     const float* __restrict__ nrm,
                                     const float* __restrict__ X,
                                     float* __restrict__ out,
                                     long long EQ, int shift4, int mask4, int shift,
                                     float scale) {
  long long i = (long long)blockIdx.x * blockDim.x + threadIdx.x;
  long long stride = (long long)gridDim.x * blockDim.x;
  for (; i < EQ; i += stride) {
    long long e = i >> shift4;
    int f = (int)(i & mask4) << 2;
    float nv = nrm[e] * scale;
    long long r = row[e];
    long long c = col[e];
    const float4 xv = *(const float4*)(X + (c << shift) + f);
    float* o = out + (r << shift) + f;
    atomicAdd(o + 0, nv * xv.x);
    atomicAdd(o + 1, nv * xv.y);
    atomicAdd(o + 2, nv * xv.z);
    atomicAdd(o + 3, nv * xv.w);
  }
}

// -------------------------------------------- weight pack (fragment-major B)
// packB linear layout: (((t*tilesN + nt)*KC + kc)*32 + lane)*2 + j
// holding W[t][kc*4 + (lane>>4)*2 + j][nt*16 + (lane&15)], zero-padded cols.

__global__ void pack_w_kernel(const float* __restrict__ W, float* __restrict__ packB,
                              int F, int Hout, int tilesN) {
  int KC = F >> 2;
  int total = 3 * tilesN * KC * 64;
  int i = blockIdx.x * blockDim.x + threadIdx.x;
  if (i >= total) return;
  int j    = i & 1;
  int lane = (i >> 1) & 31;
  int rest = i >> 6;
  int kc   = rest % KC;
  int nt   = (rest / KC) % tilesN;
  int t    = rest / (KC * tilesN);
  int k    = (kc << 2) + ((lane >> 4) << 1) + j;
  int colc = (nt << 4) + (lane & 15);
  packB[i] = (colc < Hout) ? W[((size_t)t * F + k) * Hout + colc] : 0.0f;
}

// ------------------------------------------------- fused Chebyshev GEMM (WMMA)
// out = relu?( T0@W[0] + T1@W[1] + T2@W[2] + bias ), output padded to HoutPad.
// One wave per 16x16 tile; inner loop is guard-free: 2x global_load_b64 + wmma.
// A (16x4 f32): lane L -> row (L&15), K-pair (L>=16 ? {2,3} : {0,1})
// B pre-packed so each lane loads its v2f fragment contiguously.
// C/D: VGPR r -> row r (lanes 0-15) / r+8 (lanes 16-31), col = lane&15.

template <int F, int RELU>
__global__ void cheb_gemm_kernel(const float* __restrict__ T0,
                                 const float* __restrict__ T1,
                                 const float* __restrict__ T2,
                                 const float* __restrict__ packB,
                                 const float* __restrict__ bias,  // (Hout)
                                 float* __restrict__ out,         // (Nrows, HoutPad)
                                 int Nrows, int Hout, int HoutPad, int tilesN) {
  constexpr int KC = F / 4;
  int lane = threadIdx.x & 31;
  int wave = threadIdx.x >> 5;
  int tile = blockIdx.x * (blockDim.x >> 5) + wave;
  int tilesM = Nrows >> 4;                 // Nrows is a multiple of 16
  if (tile >= tilesM * tilesN) return;     // wave-uniform exit: EXEC stays full

  int m0 = (tile / tilesN) << 4;
  int nt = tile % tilesN;

  int l15  = lane & 15;
  int kOff = (lane >> 4) << 1;             // 0 for lanes 0-15, 2 for lanes 16-31

  int colc = (nt << 4) + l15;
  float bv = (colc < Hout) ? bias[colc] : 0.0f;
  v8f acc;
#pragma unroll
  for (int r = 0; r < 8; ++r) acc[r] = bv;

  const float* terms[3] = {T0, T1, T2};
#pragma unroll
  for (int t = 0; t < 3; ++t) {
    const float* __restrict__ Ap = terms[t] + (size_t)(m0 + l15) * F + kOff;
    const float* __restrict__ Bp = packB + ((size_t)(t * tilesN + nt) * KC) * 64 + lane * 2;
#pragma unroll 8
    for (int kc = 0; kc < KC; ++kc) {
      v2f a = *(const v2f*)(Ap + (kc << 2));
      v2f b = *(const v2f*)(Bp + (kc << 6));
      // 8 args: (neg_a, A, neg_b, B, c_mod, C, reuse_a, reuse_b)
      acc = __builtin_amdgcn_wmma_f32_16x16x4_f32(
          false, a, false, b, (short)0, acc, false, false);
    }
  }

  int rowBase = m0 + ((lane >> 4) << 3);
  float* __restrict__ op = out + (size_t)rowBase * HoutPad + (nt << 4) + l15;
#pragma unroll
  for (int r = 0; r < 8; ++r) {
    float v = acc[r];
    if (RELU) v = fmaxf(v, 0.0f);
    op[(size_t)r * HoutPad] = v;
  }
}

// ------------------------------------------------------------- log-softmax
// reads padded logits (N x 16, valid cols [0,C)), writes compact (N x C)

__global__ void log_softmax_kernel(const float* __restrict__ in,
                                   float* __restrict__ out, int Nrows, int C) {
  int i = blockIdx.x * blockDim.x + threadIdx.x;
  if (i >= Nrows) return;
  const float* r = in + (size_t)i * 16;
  float m = r[0];
  for (int j = 1; j < C; ++j) m = fmaxf(m, r[j]);
  float s = 0.0f;
  for (int j = 0; j < C; ++j) s += expf(r[j] - m);
  float lse = m + logf(s);
  float* o = out + (size_t)i * C;
  for (int j = 0; j < C; ++j) o[j] = r[j] - lse;
}

// ---------------------------------------------------------------- launcher

extern "C" void kernel_launch(void* const* d_in, const int* in_sizes, int n_in,
                              void* d_out, int out_size, void* d_ws, size_t ws_size,
                              hipStream_t stream) {
  (void)n_in; (void)out_size; (void)ws_size;

  const float*     x  = (const float*)d_in[0];
  const long long* ei = (const long long*)d_in[1];   // int64 edge_index (2, E)
  const float*     ew = (const float*)d_in[2];
  const float*     W1 = (const float*)d_in[3];
  const float*     b1 = (const float*)d_in[4];
  const float*     W3 = (const float*)d_in[5];
  const float*     b3 = (const float*)d_in[6];
  const float*     W4 = (const float*)d_in[7];
  const float*     b4 = (const float*)d_in[8];
  float*           out = (float*)d_out;

  const int D = 128, H = 64, C = 5;
  const int N = in_sizes[0] / D;
  const int E = in_sizes[2];
  const long long* row = ei;
  const long long* col = ei + E;

  float* ws = (float*)d_ws;
  size_t off = 0;
  float* deg  = ws + off; off += (size_t)N;
  float* nrm  = ws + off; off += (size_t)E;
  float* T1   = ws + off; off += (size_t)N * D;      // reused across layers
  float* T2   = ws + off; off += (size_t)N * D;      // reused across layers
  float* h1   = ws + off; off += (size_t)N * H;
  float* h2   = ws + off; off += (size_t)N * H;
  float* lgts = ws + off; off += (size_t)N * 16;     // padded layer-3 logits
  float* pB1  = ws + off; off += (size_t)3 * (H / 16) * (D / 4) * 64;
  float* pB2  = ws + off; off += (size_t)3 * (H / 16) * (H / 4) * 64;
  float* pB3  = ws + off; off += (size_t)3 * 1 * (H / 4) * 64;

  dim3 blk(256);
  auto nb = [](long long n) { return dim3((unsigned)((n + 255) / 256)); };

  // ---- weight packing (tiny)
  pack_w_kernel<<<nb(3LL * (H / 16) * (D / 4) * 64), blk, 0, stream>>>(W1, pB1, D, H, H / 16);
  pack_w_kernel<<<nb(3LL * (H / 16) * (H / 4) * 64), blk, 0, stream>>>(W3, pB2, H, H, H / 16);
  pack_w_kernel<<<nb(3LL * 1 * (H / 4) * 64), blk, 0, stream>>>(W4, pB3, H, C, 1);

  // ---- normalization: deg, norm
  zero_f32_kernel<<<nb(N), blk, 0, stream>>>(deg, (size_t)N);
  deg_scatter_kernel<<<nb(E), blk, 0, stream>>>(row, ew, deg, E);
  norm_kernel<<<nb(E), blk, 0, stream>>>(row, col, ew, deg, nrm, E);

  // ---- layer 1: F = 128 -> H = 64, relu
  {
    const int F = D, shift = 7, shift4 = 5;
    const long long EQ = (long long)E << shift4;
    zero_f32_kernel<<<nb((long long)N * F), blk, 0, stream>>>(T1, (size_t)N * F);
    spmm_scatter4_kernel<<<nb(EQ), blk, 0, stream>>>(row, col, nrm, x, T1, EQ, shift4, (1 << shift4) - 1, shift, 1.0f);
    negcopy_kernel<<<nb((long long)N * F), blk, 0, stream>>>(T2, x, (size_t)N * F);
    spmm_scatter4_kernel<<<nb(EQ), blk, 0, stream>>>(row, col, nrm, T1, T2, EQ, shift4, (1 << shift4) - 1, shift, 2.0f);
    const int tilesN = H / 16;
    const long long tiles = (long long)(N / 16) * tilesN;
    cheb_gemm_kernel<128, 1><<<dim3((unsigned)((tiles + 7) / 8)), blk, 0, stream>>>(
        x, T1, T2, pB1, b1, h1, N, H, H, tilesN);
  }

  // ---- layer 2: F = 64 -> H = 64, relu
  {
    const int F = H, shift = 6, shift4 = 4;
    const long long EQ = (long long)E << shift4;
    zero_f32_kernel<<<nb((long long)N * F), blk, 0, stream>>>(T1, (size_t)N * F);
    spmm_scatter4_kernel<<<nb(EQ), blk, 0, stream>>>(row, col, nrm, h1, T1, EQ, shift4, (1 << shift4) - 1, shift, 1.0f);
    negcopy_kernel<<<nb((long long)N * F), blk, 0, stream>>>(T2, h1, (size_t)N * F);
    spmm_scatter4_kernel<<<nb(EQ), blk, 0, stream>>>(row, col, nrm, T1, T2, EQ, shift4, (1 << shift4) - 1, shift, 2.0f);
    const int tilesN = H / 16;
    const long long tiles = (long long)(N / 16) * tilesN;
    cheb_gemm_kernel<64, 1><<<dim3((unsigned)((tiles + 7) / 8)), blk, 0, stream>>>(
        h1, T1, T2, pB2, b3, h2, N, H, H, tilesN);
  }

  // ---- layer 3: F = 64 -> C = 5 (zero-padded to one 16-wide tile), no relu
  {
    const int F = H, shift = 6, shift4 = 4;
    const long long EQ = (long long)E << shift4;
    zero_f32_kernel<<<nb((long long)N * F), blk, 0, stream>>>(T1, (size_t)N * F);
    spmm_scatter4_kernel<<<nb(EQ), blk, 0, stream>>>(row, col, nrm, h2, T1, EQ, shift4, (1 << shift4) - 1, shift, 1.0f);
    negcopy_kernel<<<nb((long long)N * F), blk, 0, stream>>>(T2, h2, (size_t)N * F);
    spmm_scatter4_kernel<<<nb(EQ), blk, 0, stream>>>(row, col, nrm, T1, T2, EQ, shift4, (1 << shift4) - 1, shift, 2.0f);
    const long long tiles = (long long)(N / 16);
    cheb_gemm_kernel<64, 0><<<dim3((unsigned)((tiles + 7) / 8)), blk, 0, stream>>>(
        h2, T1, T2, pB3, b4, lgts, N, C, 16, 1);
  }

  // ---- log_softmax over C=5 (read padded, write compact)
  log_softmax_kernel<<<nb(N), blk, 0, stream>>>(lgts, out, N, C);
}